// FlatSE3AttentionHead_11656541241399
// MI455X (gfx1250) — compile-verified
//
#include <hip/hip_runtime.h>
#include <math.h>

// CDNA5 / gfx1250: wave32, WMMA 16x16x4 f32.
typedef __attribute__((ext_vector_type(2))) float v2f;   // A/B operand: 16x4 / 4x16 f32 = 2 VGPRs/lane
typedef __attribute__((ext_vector_type(8))) float v8f;   // C/D operand: 16x16 f32 = 8 VGPRs/lane

#define B_N 4096
#define INV_SQRT3 0.57735026918962576f
#define LOG2E     1.4426950408889634f
// K pre-scale: fold both the attention scale (3^-1/2) and the -log2(e) needed by
// the native exp2 unit into K, so sigmoid(x) = rcp(1 + exp2(wmma_output)).
#define K_SCALE   (-LOG2E * INV_SQRT3)

// ---------------- Stage 1: per-point projections into d_ws ----------------
// ws layout: 8 arrays of B_N float4:
//   [0]=Qx {q,0}  [1]=Kx {k * K_SCALE, 0}  [2]=Vx {v,0}
//   [3]=Qy        [4]=Ky (scaled)          [5]=Vy
//   [6]=Tm {t.x,t.y,t.z,|t|^2}             [7]=Vt {v,0}

__device__ __forceinline__ void lin3(const float* __restrict__ w,
                                     const float* __restrict__ b,
                                     float a0, float a1, float a2, float s,
                                     float4& o) {
    o.x = (fmaf(w[0], a0, fmaf(w[1], a1, fmaf(w[2], a2, b[0])))) * s;
    o.y = (fmaf(w[3], a0, fmaf(w[4], a1, fmaf(w[5], a2, b[1])))) * s;
    o.z = (fmaf(w[6], a0, fmaf(w[7], a1, fmaf(w[8], a2, b[2])))) * s;
    o.w = 0.0f;
}

__global__ void se3_proj_kernel(
    const float* __restrict__ tf,
    const float* __restrict__ qxw, const float* __restrict__ qxb,
    const float* __restrict__ kxw, const float* __restrict__ kxb,
    const float* __restrict__ vxw, const float* __restrict__ vxb,
    const float* __restrict__ qyw, const float* __restrict__ qyb,
    const float* __restrict__ kyw, const float* __restrict__ kyb,
    const float* __restrict__ vyw, const float* __restrict__ vyb,
    const float* __restrict__ vtw, const float* __restrict__ vtb,
    float4* __restrict__ ws)
{
    int i = blockIdx.x * blockDim.x + threadIdx.x;
    if (i >= B_N) return;
    const float* r = tf + i * 16;               // transforms[i] is 4x4 row-major
    float x0 = r[0], x1 = r[4], x2 = r[8];      // transforms[:, :3, 0]
    float t0 = r[3], t1 = r[7], t2 = r[11];     // transforms[:, :3, 3]

    float4 o;
    lin3(qxw, qxb, x0, x1, x2, 1.0f,    o); ws[0 * B_N + i] = o;
    lin3(kxw, kxb, x0, x1, x2, K_SCALE, o); ws[1 * B_N + i] = o;  // fold 3^-1/2 * -log2(e)
    lin3(vxw, vxb, x0, x1, x2, 1.0f,    o); ws[2 * B_N + i] = o;
    lin3(qyw, qyb, x0, x1, x2, 1.0f,    o); ws[3 * B_N + i] = o;  // yr == xr
    lin3(kyw, kyb, x0, x1, x2, K_SCALE, o); ws[4 * B_N + i] = o;
    lin3(vyw, vyb, x0, x1, x2, 1.0f,    o); ws[5 * B_N + i] = o;
    float4 tm; tm.x = t0; tm.y = t1; tm.z = t2;
    tm.w = fmaf(t0, t0, fmaf(t1, t1, t2 * t2));
    ws[6 * B_N + i] = tm;
    lin3(vtw, vtb, t0, t1, t2, 1.0f,    o); ws[7 * B_N + i] = o;
}

// ---------------- Stage 2: fused 3-head attention, WMMA logits ----------------
// One wave per 16-row i-tile (256 waves total). Per 16-wide j-tile:
//   dx = Qx(16x4) x Kx'^T(4x16)           -> -log2(e)*logit; sigmoid = rcp(1+exp2(.))
//   dy = Qy(16x4) x Ky'^T(4x16)
//   dt = {-2t_i,1}(16x4) x {t_j,|t_j|^2}^T(4x16) + C{|t_i|^2}  =  |t_i - t_j|^2
//        (|t_i|^2 rides in the WMMA C operand: per-row constant = exactly C layout)
// Tiles stay in WMMA D layout: lane holds column j0+(lane&15), rows r+8*(lane>=16).
// Each lane accumulates att*v partials for its j-slice; one shfl_xor butterfly at
// the end reduces the 16-lane halves into full row sums.

__global__ void __launch_bounds__(256) se3_attn_kernel(const float4* __restrict__ ws,
                                                       float* __restrict__ out)
{
    const float4* Qx = ws + 0 * B_N;
    const float4* Kx = ws + 1 * B_N;
    const float4* Vx = ws + 2 * B_N;
    const float4* Qy = ws + 3 * B_N;
    const float4* Ky = ws + 4 * B_N;
    const float4* Vy = ws + 5 * B_N;
    const float4* Tm = ws + 6 * B_N;
    const float4* Vt = ws + 7 * B_N;

    const int lane = threadIdx.x & 31;
    const int half = lane >> 4;          // 0: K/M-low half, 1: K/M-high half
    const int lm   = lane & 15;
    const int wave = (blockIdx.x * blockDim.x + threadIdx.x) >> 5;
    const int i0   = wave * 16;

    // A operands (fixed per wave). 32-bit A 16x4: lanes 0-15 hold K={0,1} of row lm,
    // lanes 16-31 hold K={2,3}.
    const int arow = i0 + lm;
    float4 qx = Qx[arow], qy = Qy[arow], tm = Tm[arow];
    v2f a_x, a_y, a_t;
    if (half) { a_x[0] = qx.z; a_x[1] = qx.w;
                a_y[0] = qy.z; a_y[1] = qy.w;
                a_t[0] = -2.0f * tm.z; a_t[1] = 1.0f; }
    else      { a_x[0] = qx.x; a_x[1] = qx.y;
                a_y[0] = qy.x; a_y[1] = qy.y;
                a_t[0] = -2.0f * tm.x; a_t[1] = -2.0f * tm.y; }

    // |t_i|^2 for this lane's 8 output rows (M = r + 8*half) -> WMMA C operand.
    v8f cn;
    #pragma unroll
    for (int r = 0; r < 8; ++r) cn[r] = Tm[i0 + r + 8 * half].w;

    float ax_[8][3] = {}, ay_[8][3] = {}, st_[8][3] = {};
    float dmax[8];
    #pragma unroll
    for (int r = 0; r < 8; ++r) dmax[r] = 0.0f;
    float vs0 = 0.0f, vs1 = 0.0f, vs2 = 0.0f;   // sum_j vt_j (per j-slice)

    const v2f* Kx2 = (const v2f*)Kx;
    const v2f* Ky2 = (const v2f*)Ky;
    const v2f* Tm2 = (const v2f*)Tm;

    for (int j0 = 0; j0 < B_N; j0 += 16) {
        const int jb = j0 + lm;
        // B 4x16: lanes 0-15 hold K={0,1} of column lm, lanes 16-31 hold K={2,3}.
        v2f b_x = Kx2[2 * jb + half];
        v2f b_y = Ky2[2 * jb + half];
        v2f b_t = Tm2[2 * jb + half];   // half==1 slot K=3 carries |t_j|^2, A slot = 1.0
        float4 vx = Vx[jb], vy = Vy[jb], vt = Vt[jb];

        v8f cz = {};
        v8f dx = __builtin_amdgcn_wmma_f32_16x16x4_f32(false, a_x, false, b_x, (short)0, cz, false, false);
        v8f dy = __builtin_amdgcn_wmma_f32_16x16x4_f32(false, a_y, false, b_y, (short)0, cz, false, false);
        v8f dt = __builtin_amdgcn_wmma_f32_16x16x4_f32(false, a_t, false, b_t, (short)0, cn, false, false);

        #pragma unroll
        for (int r = 0; r < 8; ++r) {
            // sigmoid: K carries -log2(e)/sqrt(3), so this is rcp(1 + 2^(-x*log2e))
            float px = __builtin_amdgcn_rcpf(1.0f + __builtin_amdgcn_exp2f(dx[r]));
            ax_[r][0] = fmaf(px, vx.x, ax_[r][0]);
            ax_[r][1] = fmaf(px, vx.y, ax_[r][1]);
            ax_[r][2] = fmaf(px, vx.z, ax_[r][2]);
            float py = __builtin_amdgcn_rcpf(1.0f + __builtin_amdgcn_exp2f(dy[r]));
            ay_[r][0] = fmaf(py, vy.x, ay_[r][0]);
            ay_[r][1] = fmaf(py, vy.y, ay_[r][1]);
            ay_[r][2] = fmaf(py, vy.z, ay_[r][2]);
            // dt[r] already = |t_i - t_j|^2 (C operand carried |t_i|^2)
            float d  = __builtin_amdgcn_sqrtf(fmaxf(dt[r], 0.0f));  // raw v_sqrt_f32
            dmax[r]  = fmaxf(dmax[r], d);
            st_[r][0] = fmaf(d, vt.x, st_[r][0]);
            st_[r][1] = fmaf(d, vt.y, st_[r][1]);
            st_[r][2] = fmaf(d, vt.z, st_[r][2]);
        }
        vs0 += vt.x; vs1 += vt.y; vs2 += vt.z;
    }

    // Butterfly-reduce across the 16 lanes of each half (masks 1..8 never cross bit 4).
    #pragma unroll
    for (int m = 1; m < 16; m <<= 1) {
        #pragma unroll
        for (int r = 0; r < 8; ++r) {
            #pragma unroll
            for (int c = 0; c < 3; ++c) {
                ax_[r][c] += __shfl_xor(ax_[r][c], m, 32);
                ay_[r][c] += __shfl_xor(ay_[r][c], m, 32);
                st_[r][c] += __shfl_xor(st_[r][c], m, 32);
            }
            dmax[r] = fmaxf(dmax[r], __shfl_xor(dmax[r], m, 32));
        }
        vs0 += __shfl_xor(vs0, m, 32);
        vs1 += __shfl_xor(vs1, m, 32);
        vs2 += __shfl_xor(vs2, m, 32);
    }

    // Lane lm==r of each half writes row i0 + r + 8*half.
    // Output columns: [0:3]=att_t@tr_v  [3:6]=att_y@yr_v  [6:9]=att_x@xr_v
    #pragma unroll
    for (int r = 0; r < 8; ++r) {
        if (lm == r) {
            const int row = i0 + r + 8 * half;
            const float inv = 1.0f / dmax[r];   // once per row: keep precise divide
            float* o = out + row * 9;
            o[0] = fmaf(2.0f, vs0, -st_[r][0] * inv);
            o[1] = fmaf(2.0f, vs1, -st_[r][1] * inv);
            o[2] = fmaf(2.0f, vs2, -st_[r][2] * inv);
            o[3] = ay_[r][0]; o[4] = ay_[r][1]; o[5] = ay_[r][2];
            o[6] = ax_[r][0]; o[7] = ax_[r][1]; o[8] = ax_[r][2];
        }
    }
}

// ---------------- launch ----------------
extern "C" void kernel_launch(void* const* d_in, const int* in_sizes, int n_in,
                              void* d_out, int out_size, void* d_ws, size_t ws_size,
                              hipStream_t stream) {
    const float* tf = (const float*)d_in[0];
    float4* ws = (float4*)d_ws;   // needs 8 * 4096 * 16 B = 512 KB

    se3_proj_kernel<<<B_N / 256, 256, 0, stream>>>(
        tf,
        (const float*)d_in[1],  (const float*)d_in[2],
        (const float*)d_in[3],  (const float*)d_in[4],
        (const float*)d_in[5],  (const float*)d_in[6],
        (const float*)d_in[7],  (const float*)d_in[8],
        (const float*)d_in[9],  (const float*)d_in[10],
        (const float*)d_in[11], (const float*)d_in[12],
        (const float*)d_in[13], (const float*)d_in[14],
        ws);

    // 256 waves (one per 16-row tile): 32 blocks x 256 threads (8 waves/block).
    se3_attn_kernel<<<32, 256, 0, stream>>>((const float4*)d_ws, (float*)d_out);
}